// GANLoss_19155554140635
// MI455X (gfx1250) — compile-verified
//
#include <hip/hip_runtime.h>
#include <hip/hip_bf16.h>

typedef __attribute__((ext_vector_type(2))) float v2f;
typedef __attribute__((ext_vector_type(8))) float v8f;

// Problem constants from the reference (fixed shapes).
#define GB 16       // BATCH_SIZE
#define GS 512      // g_sequence_len
#define GV 32000    // VOCAB_SIZE
#define NTHREADS 1024               // 32 wave32 waves on one WGP
#define PER_THREAD ((GB * GS) / NTHREADS)  // 8 gathers per thread

__global__ __launch_bounds__(NTHREADS)
void gan_loss_kernel(const float* __restrict__ prob,
                     const int*   __restrict__ samples,
                     const float* __restrict__ reward,
                     const float* __restrict__ c_phi_z,
                     const float* __restrict__ c_phi_z_tilde,
                     float* __restrict__ out)
{
    __shared__ float wave_partial[32];

    const int tid = threadIdx.x;

    // ---- Gather + weighted accumulate: 8 independent loads per thread ----
    float acc = 0.0f;
#pragma unroll
    for (int r = 0; r < PER_THREAD; ++r) {
        const int i = tid + r * NTHREADS;        // 0 .. B*S-1
        const int b = i >> 9;                    // i / GS  (GS = 512)
        const int idx = samples[i];              // vocab index
        const float w = reward[b] - c_phi_z_tilde[2 * b + 1];
        acc += prob[(size_t)i * (size_t)GV + (size_t)idx] * w;
    }

    // Fold the per-batch constant term  (cz[b] - czt[b])  into threads 0..15.
    if (tid < GB)
        acc += c_phi_z[2 * tid + 1] - c_phi_z_tilde[2 * tid + 1];

    // ---- Intra-wave reduction (wave32) ----
#pragma unroll
    for (int off = 16; off > 0; off >>= 1)
        acc += __shfl_xor(acc, off, 32);

    const int wave = tid >> 5;
    const int lane = tid & 31;
    if (lane == 0)
        wave_partial[wave] = acc;
    __syncthreads();

    // ---- Cross-wave reduction via V_WMMA_F32_16X16X4_F32 (wave 0 only) ----
    // A (16x4 f32, 2 VGPRs): lane w, VGPR0 -> A[w%16, (w<16)?0:2] = partial_w,
    // VGPR1 -> 0.  B = all ones.  D[m,n] = partial_m + partial_{m+16}.
    // Lane 0 holds D[0..7,0] in c[0..7]; lane 16 holds D[8..15,0].
    if (tid < 32) {   // wave-uniform guard: wave 0 has EXEC = all ones
        v2f a;
        a.x = wave_partial[tid];
        a.y = 0.0f;
        v2f bm;
        bm.x = 1.0f;
        bm.y = 1.0f;
        v8f c = {};
        c = __builtin_amdgcn_wmma_f32_16x16x4_f32(
                /*neg_a=*/false, a,
                /*neg_b=*/false, bm,
                /*c_mod=*/(short)0, c,
                /*reuse_a=*/false, /*reuse_b=*/false);

        float s = c[0] + c[1] + c[2] + c[3] + c[4] + c[5] + c[6] + c[7];
        // total = (rows 0..7 sum, in lane 0) + (rows 8..15 sum, in lane 16)
        float total = __shfl(s, 0, 32) + __shfl(s, 16, 32);
        if (tid == 0)
            *out = -total;
    }
}

extern "C" void kernel_launch(void* const* d_in, const int* in_sizes, int n_in,
                              void* d_out, int out_size, void* d_ws, size_t ws_size,
                              hipStream_t stream)
{
    (void)in_sizes; (void)n_in; (void)d_ws; (void)ws_size; (void)out_size;

    const float* prob          = (const float*)d_in[0];
    const int*   samples       = (const int*)  d_in[1];
    const float* reward        = (const float*)d_in[2];
    const float* c_phi_z       = (const float*)d_in[3];
    const float* c_phi_z_tilde = (const float*)d_in[4];
    float* out = (float*)d_out;

    gan_loss_kernel<<<1, NTHREADS, 0, stream>>>(prob, samples, reward,
                                                c_phi_z, c_phi_z_tilde, out);
}